// sparse_cat_fuse_45964740001830
// MI455X (gfx1250) — compile-verified
//
#include <hip/hip_runtime.h>
#include <stdint.h>

// ---------------- constants -----------------------------------------------
#define CHN        32
#define CAPLOG     21
#define HCAP       (1u << CAPLOG)          // hash table slots (2.1M for 800K keys)
#define HEMPTY     0xFFFFFFFFu
#define DENSE_BITS (164u * 1024u * 1024u)  // 4*41*1024*1024 dense coord space
#define NWORDS     (DENSE_BITS / 32u)      // 5,373,952 bitmap words
#define SCAN_WPB   2048                    // bitmap words per scan block (8KB)
#define TROWS      64                      // teacher rows per block (8/wave)

// dense key: lexicographic (b,z,y,x) -> same ordering as the int64 hash
__device__ __forceinline__ uint32_t dense_key(int b, int z, int y, int x) {
    return (((uint32_t)b * 41u + (uint32_t)z) << 20) |
           ((uint32_t)y << 10) | (uint32_t)x;
}

__device__ __forceinline__ uint32_t tab_hash(uint32_t key) {
    return (key * 2654435761u) >> (32 - CAPLOG);
}

// ---------------- TDM: 1-D tensor_load_to_lds (CDNA5) ----------------------
typedef unsigned int u32x4 __attribute__((ext_vector_type(4)));
typedef int          i32x8 __attribute__((ext_vector_type(8)));
typedef int          i32x4 __attribute__((ext_vector_type(4)));

// Copy `tile_dwords` DWORDs from gptr to LDS offset lds_off; reads past
// `tensor_dwords` return zero (TDM OOB semantics).
__device__ __forceinline__ void tdm_load_1d(uint32_t lds_off, const void* gptr,
                                            uint32_t tile_dwords,
                                            uint32_t tensor_dwords) {
    uint64_t ga = (uint64_t)(uintptr_t)gptr;
    u32x4 g0;
    g0[0] = 1u;                                   // count=1, user desc
    g0[1] = lds_off;                              // lds_addr (bytes)
    g0[2] = (uint32_t)ga;                         // global_addr[31:0]
    g0[3] = (uint32_t)(ga >> 32) | (2u << 30);    // global_addr[56:32] | type=2
    i32x8 g1;
    g1[0] = (int)(2u << 16);                      // data_size = 4B, no mcast/pad
    g1[1] = (int)((tensor_dwords & 0xFFFFu) << 16);        // tensor_dim0 lo16
    g1[2] = (int)((tensor_dwords >> 16) & 0xFFFFu);        // tensor_dim0 hi16
    g1[3] = (int)((tile_dwords & 0xFFFFu) << 16);          // tile_dim0
    g1[4] = 0;                                    // tile_dim1=0 -> 1D
    g1[5] = 0; g1[6] = 0; g1[7] = 0;              // strides unused for 1D
    i32x4 g2 = {0, 0, 0, 0};
    i32x4 g3 = {0, 0, 0, 0};
    i32x8 gx = {0, 0, 0, 0, 0, 0, 0, 0};          // 6-arg toolchain: extra group
    __builtin_amdgcn_tensor_load_to_lds(g0, g1, g2, g3, gx, 0);
}

// ---------------- K1: student bitmap + hash-table insert --------------------
__global__ void k_student_insert(const int* __restrict__ sidx,
                                 uint32_t* __restrict__ bitmap,
                                 uint32_t* __restrict__ keys,
                                 uint32_t* __restrict__ val, int NS) {
    int s = blockIdx.x * blockDim.x + threadIdx.x;
    if (s >= NS) return;
    int4 c = ((const int4*)sidx)[s];
    uint32_t key = dense_key(c.x, c.y, c.z, c.w);
    atomicOr(&bitmap[key >> 5], 1u << (key & 31));
    uint32_t h = tab_hash(key);
    for (;;) {
        uint32_t prev = atomicCAS(&keys[h], HEMPTY, key);
        if (prev == HEMPTY) { val[h] = (uint32_t)s; break; }   // keys are unique
        h = (h + 1) & (HCAP - 1);
    }
}

// ---------------- K2a: per-block popcount sums (TDM-staged) -----------------
__global__ void k_popcount_blocksum(const uint32_t* __restrict__ bitmap,
                                    uint32_t* __restrict__ blockSums) {
    __shared__ uint32_t wbuf[SCAN_WPB];
    __shared__ uint32_t red[256];
    uint32_t base = blockIdx.x * SCAN_WPB;
    if (threadIdx.x < 32) {
        tdm_load_1d((uint32_t)(uintptr_t)&wbuf[0], bitmap + base,
                    SCAN_WPB, NWORDS - base);
        __builtin_amdgcn_s_wait_tensorcnt(0);
    }
    __syncthreads();
    uint32_t sum = 0;
    for (int j = threadIdx.x; j < SCAN_WPB; j += 256) sum += __popc(wbuf[j]);
    red[threadIdx.x] = sum;
    __syncthreads();
    for (int off = 128; off > 0; off >>= 1) {
        if (threadIdx.x < (unsigned)off) red[threadIdx.x] += red[threadIdx.x + off];
        __syncthreads();
    }
    if (threadIdx.x == 0) blockSums[blockIdx.x] = red[0];
}

// ---------------- K2b: single-block exclusive scan of block sums ------------
__global__ void k_scan_blocksums(uint32_t* __restrict__ bs, int nb) {
    __shared__ uint32_t sh[256];
    uint32_t carry = 0;
    for (int base = 0; base < nb; base += 256) {
        int i = base + (int)threadIdx.x;
        uint32_t v = (i < nb) ? bs[i] : 0u;
        sh[threadIdx.x] = v;
        __syncthreads();
        for (int off = 1; off < 256; off <<= 1) {
            uint32_t x = (threadIdx.x >= (unsigned)off) ? sh[threadIdx.x - off] : 0u;
            __syncthreads();
            sh[threadIdx.x] += x;
            __syncthreads();
        }
        uint32_t incl = sh[threadIdx.x];
        uint32_t tot  = sh[255];
        if (i < nb) bs[i] = carry + incl - v;     // exclusive prefix
        carry += tot;
        __syncthreads();
    }
}

// ---------------- K2c: exclusive bit-prefix per bitmap word (TDM-staged) ----
__global__ void k_write_wordprefix(const uint32_t* __restrict__ bitmap,
                                   const uint32_t* __restrict__ blockSums,
                                   uint32_t* __restrict__ wordPrefix) {
    __shared__ uint32_t wbuf[SCAN_WPB];
    __shared__ uint32_t sh[256];
    uint32_t base = blockIdx.x * SCAN_WPB;
    if (threadIdx.x < 32) {
        tdm_load_1d((uint32_t)(uintptr_t)&wbuf[0], bitmap + base,
                    SCAN_WPB, NWORDS - base);
        __builtin_amdgcn_s_wait_tensorcnt(0);
    }
    __syncthreads();
    uint32_t loc[8], s8 = 0;
    for (int k = 0; k < 8; ++k) {
        loc[k] = (uint32_t)__popc(wbuf[threadIdx.x * 8 + k]);
        s8 += loc[k];
    }
    sh[threadIdx.x] = s8;
    __syncthreads();
    for (int off = 1; off < 256; off <<= 1) {
        uint32_t x = (threadIdx.x >= (unsigned)off) ? sh[threadIdx.x - off] : 0u;
        __syncthreads();
        sh[threadIdx.x] += x;
        __syncthreads();
    }
    uint32_t run = blockSums[blockIdx.x] + sh[threadIdx.x] - s8;  // exclusive
    for (int k = 0; k < 8; ++k) {
        uint32_t w = base + threadIdx.x * 8 + k;
        if (w < NWORDS) wordPrefix[w] = run;
        run += loc[k];
    }
}

// ---------------- K3: rank of each student key (== sorted output row) -------
__global__ void k_rank(const int* __restrict__ sidx,
                       const uint32_t* __restrict__ bitmap,
                       const uint32_t* __restrict__ wordPrefix,
                       uint32_t* __restrict__ rankOf, int NS) {
    int s = blockIdx.x * blockDim.x + threadIdx.x;
    if (s >= NS) return;
    int4 c = ((const int4*)sidx)[s];
    uint32_t key = dense_key(c.x, c.y, c.z, c.w);
    uint32_t w = key >> 5, b = key & 31;
    uint32_t mask = (1u << b) - 1u;                 // b==0 -> 0
    rankOf[s] = wordPrefix[w] + (uint32_t)__popc(bitmap[w] & mask);
}

// ---------------- K4: teacher pass — wave32 per row, 8 rows/wave ------------
// Lanes == channels. Features TDM-staged (64 rows x 32ch = 8KB/block); the
// DMA overlaps the hash probes, which don't touch feature data.
__global__ void k_teacher(const float* __restrict__ tf,
                          const int* __restrict__ ti,
                          const uint32_t* __restrict__ keys,
                          const uint32_t* __restrict__ val,
                          const uint32_t* __restrict__ rankOf,
                          int* __restrict__ firstIdx,
                          float* __restrict__ outFeat, int NT) {
    __shared__ float fsh[TROWS * CHN];
    const int t0 = blockIdx.x * TROWS;
    if (threadIdx.x < 32) {                       // wave 0 issues the DMA
        uint64_t rem64 = (uint64_t)NT * CHN - (uint64_t)t0 * CHN;
        uint32_t rem = (rem64 > 0xFFFFFFFFull) ? 0xFFFFFFFFu : (uint32_t)rem64;
        tdm_load_1d((uint32_t)(uintptr_t)&fsh[0], tf + (size_t)t0 * CHN,
                    TROWS * CHN, rem);
        // no wait yet: overlap DMA with the probe phase below
    }
    __builtin_prefetch(ti + (size_t)(t0 + TROWS) * 4, 0, 0); // global_prefetch_b8

    const int wave = threadIdx.x >> 5, lane = threadIdx.x & 31;
    uint32_t srow[8];                             // matched student id per row
    uint32_t rrow[8];                             // output rank per row
    bool     hit[8];
    for (int q = 0; q < 8; ++q) {
        int t = t0 + wave * 8 + q;
        hit[q] = false;
        if (t < NT) {
            // wave-uniform address -> scalarizable loads & uniform probe
            int b = ti[t * 4 + 0], z = ti[t * 4 + 1];
            int y = ti[t * 4 + 2], x = ti[t * 4 + 3];
            uint32_t key = dense_key(b, z, y, x);
            uint32_t h = tab_hash(key);
            for (;;) {
                uint32_t k = keys[h];
                if (k == HEMPTY) break;           // not a student coord: discard
                if (k == key) {
                    uint32_t s = val[h];
                    srow[q] = s;
                    rrow[q] = rankOf[s];
                    hit[q]  = true;
                    break;
                }
                h = (h + 1) & (HCAP - 1);
            }
        }
    }
    if (threadIdx.x < 32) __builtin_amdgcn_s_wait_tensorcnt(0);
    __syncthreads();                              // features visible in LDS

    for (int q = 0; q < 8; ++q) {
        if (!hit[q]) continue;
        int t = t0 + wave * 8 + q;
        if (lane == 0) atomicMin(&firstIdx[srow[q]], t);
        atomicAdd(&outFeat[(size_t)rrow[q] * CHN + lane],
                  fsh[(wave * 8 + q) * CHN + lane]);
    }
}

// ---------------- K5: scatter first-occurrence index rows -------------------
__global__ void k_write_indices(const int* __restrict__ ti,
                                const uint32_t* __restrict__ rankOf,
                                const int* __restrict__ firstIdx,
                                int* __restrict__ outIdx, int NS) {
    int s = blockIdx.x * blockDim.x + threadIdx.x;
    if (s >= NS) return;
    uint32_t r = rankOf[s];
    int fi = firstIdx[s];
    int4 row = ((const int4*)ti)[fi];
    ((int4*)outIdx)[r] = row;
}

// ---------------- launcher --------------------------------------------------
extern "C" void kernel_launch(void* const* d_in, const int* in_sizes, int n_in,
                              void* d_out, int out_size, void* d_ws, size_t ws_size,
                              hipStream_t stream) {
    const float* tf = (const float*)d_in[0];    // [NT, 32] f32
    const int*   ti = (const int*)d_in[1];      // [NT, 4]  i32
    const int*   si = (const int*)d_in[2];      // [NS, 4]  i32
    const int NT = in_sizes[0] / CHN;
    const int NS = in_sizes[2] / 4;

    float* outFeat = (float*)d_out;                            // [NS,32] f32
    int*   outIdx  = (int*)((float*)d_out + (size_t)NS * CHN); // [NS,4] i32

    char* wsb = (char*)d_ws;
    size_t off = 0;
    auto alloc = [&](size_t bytes) -> void* {
        void* p = wsb + off;
        off = (off + bytes + 255) & ~(size_t)255;
        return p;
    };
    const int nb = (int)((NWORDS + SCAN_WPB - 1) / SCAN_WPB);
    uint32_t* bitmap     = (uint32_t*)alloc((size_t)NWORDS * 4);
    uint32_t* wordPrefix = (uint32_t*)alloc((size_t)NWORDS * 4);
    uint32_t* blockSums  = (uint32_t*)alloc((size_t)nb * 4);
    uint32_t* keys       = (uint32_t*)alloc((size_t)HCAP * 4);
    uint32_t* val        = (uint32_t*)alloc((size_t)HCAP * 4);
    uint32_t* rankOf     = (uint32_t*)alloc((size_t)NS * 4);
    int*      firstIdx   = (int*)alloc((size_t)NS * 4);

    (void)hipMemsetAsync(bitmap, 0x00, (size_t)NWORDS * 4, stream);
    (void)hipMemsetAsync(keys,   0xFF, (size_t)HCAP * 4, stream);
    (void)hipMemsetAsync(firstIdx, 0x7F, (size_t)NS * 4, stream); // +large for atomicMin
    (void)hipMemsetAsync(outFeat, 0x00, (size_t)NS * CHN * sizeof(float), stream);

    k_student_insert<<<(NS + 255) / 256, 256, 0, stream>>>(si, bitmap, keys, val, NS);
    k_popcount_blocksum<<<nb, 256, 0, stream>>>(bitmap, blockSums);
    k_scan_blocksums<<<1, 256, 0, stream>>>(blockSums, nb);
    k_write_wordprefix<<<nb, 256, 0, stream>>>(bitmap, blockSums, wordPrefix);
    k_rank<<<(NS + 255) / 256, 256, 0, stream>>>(si, bitmap, wordPrefix, rankOf, NS);
    k_teacher<<<(NT + TROWS - 1) / TROWS, 256, 0, stream>>>(tf, ti, keys, val, rankOf,
                                                            firstIdx, outFeat, NT);
    k_write_indices<<<(NS + 255) / 256, 256, 0, stream>>>(ti, rankOf, firstIdx,
                                                          outIdx, NS);
}